// OpticalConv2dNew_55413668053404
// MI455X (gfx1250) — compile-verified
//
#include <hip/hip_runtime.h>

typedef float v2f __attribute__((ext_vector_type(2)));
typedef float v8f __attribute__((ext_vector_type(8)));

#define B_SZ    8
#define IN_CH   32
#define OUT_CH  64
#define NP      64    // image side
#define NTAP    49    // 7x7 taps
#define KPAD    52    // taps padded to multiple of 4 (13 wmma k-steps)

// Workgroup: 128 threads = 4 waves. One workgroup handles (b, oTile[16], row y).
// Each wave handles 16 consecutive x-pixels -> the 4 waves cover the full row.
// Per input channel c: stage weights (16x52) + 7 wrapped image rows (7x64) in
// LDS, run 13 x V_WMMA_F32_16X16X4_F32 into a per-channel accumulator, then
// acc += |cacc| (abs is inside the channel sum in the reference).
__global__ __launch_bounds__(128) void opt_conv_wmma_f32(
    const float* __restrict__ img,   // (8,32,64,64)
    const float* __restrict__ ker,   // (64,32,7,7)
    float* __restrict__ out)         // (8,64,64,64)
{
    __shared__ float sK[16 * KPAD];  // [m][t]  weights, zero-padded taps 49..51
    __shared__ float sI[7 * NP];     // [i][x]  image rows (y+4-i)%64

    const int blk = blockIdx.x;           // 8 * 4 * 64 = 2048 blocks
    const int b   = blk >> 8;
    const int rem = blk & 255;
    const int ot  = rem >> 6;             // o-tile: o in [16*ot, 16*ot+16)
    const int y   = rem & 63;

    const int tid  = threadIdx.x;
    const int wave = tid >> 5;
    const int lane = tid & 31;
    const int x0   = wave << 4;           // this wave's 16 pixels
    const int nloc = lane & 15;           // N index within tile (pixel / M row)
    const int hi   = lane >> 4;           // half-wave select (K pair / M+8)

    v8f acc = {0.f, 0.f, 0.f, 0.f, 0.f, 0.f, 0.f, 0.f};

    for (int c = 0; c < IN_CH; ++c) {
        // ---- cooperative stage into LDS ----
        for (int idx = tid; idx < 16 * KPAD; idx += 128) {
            int m = idx / KPAD;
            int t = idx - m * KPAD;
            float v = 0.f;
            if (t < NTAP) v = ker[((ot * 16 + m) * IN_CH + c) * NTAP + t];
            sK[idx] = v;
        }
        for (int idx = tid; idx < 7 * NP; idx += 128) {
            int i   = idx >> 6;
            int x   = idx & 63;
            int row = (y + 4 - i + NP) & 63;          // wrap-around row
            sI[idx] = img[((b * IN_CH + c) * NP + row) * NP + x];
        }
        // prefetch next channel's image rows (global_prefetch_b8)
        if (c + 1 < IN_CH) {
            const float* nxt = &img[((b * IN_CH + c + 1) * NP + ((y + tid) & 63)) * NP];
            __builtin_prefetch(nxt, 0, 2);
        }
        __syncthreads();

        // ---- 16x16 fp32 WMMA tile: D = W(16x52) * im2col(52x16) ----
        v8f cacc = {0.f, 0.f, 0.f, 0.f, 0.f, 0.f, 0.f, 0.f};
        #pragma unroll
        for (int kc = 0; kc < 13; ++kc) {
            const int kb = kc * 4 + hi * 2;           // this lane's K pair base
            // A fragment: 16x4 fp32 (lane m = nloc; VGPR0=K kb, VGPR1=K kb+1)
            v2f a;
            a.x = sK[nloc * KPAD + kb];
            a.y = sK[nloc * KPAD + kb + 1];
            // B fragment: 4x16 fp32 (lane n = nloc; rows kb, kb+1)
            int t0 = kb     > 48 ? 48 : kb;           // taps >=49 have A==0
            int t1 = kb + 1 > 48 ? 48 : kb + 1;
            int i0 = t0 / 7, j0 = t0 - i0 * 7;
            int i1 = t1 / 7, j1 = t1 - i1 * 7;
            v2f bf;
            bf.x = sI[i0 * NP + ((x0 + nloc + 4 - j0 + NP) & 63)];
            bf.y = sI[i1 * NP + ((x0 + nloc + 4 - j1 + NP) & 63)];
            cacc = __builtin_amdgcn_wmma_f32_16x16x4_f32(
                /*neg_a=*/false, a, /*neg_b=*/false, bf,
                /*c_mod=*/(short)0, cacc, /*reuse_a=*/false, /*reuse_b=*/false);
        }
        #pragma unroll
        for (int r = 0; r < 8; ++r) acc[r] += fabsf(cacc[r]);
        __syncthreads();   // protect LDS before next channel's staging
    }

    // ---- write 16x16 tile: VGPR r -> M = r + 8*hi, N = nloc ----
    #pragma unroll
    for (int r = 0; r < 8; ++r) {
        int o = ot * 16 + hi * 8 + r;
        int x = x0 + nloc;
        out[((b * OUT_CH + o) * NP + y) * NP + x] = acc[r];
    }
}

extern "C" void kernel_launch(void* const* d_in, const int* in_sizes, int n_in,
                              void* d_out, int out_size, void* d_ws, size_t ws_size,
                              hipStream_t stream) {
    const float* img = (const float*)d_in[0];   // (8,32,64,64) fp32
    const float* ker = (const float*)d_in[1];   // (64,32,7,7)  fp32
    float* out       = (float*)d_out;           // (8,64,64,64) fp32
    (void)in_sizes; (void)n_in; (void)out_size; (void)d_ws; (void)ws_size;

    dim3 grid(B_SZ * 4 * NP);   // 8 batches * 4 o-tiles * 64 rows = 2048
    dim3 block(128);            // 4 waves of 32
    opt_conv_wmma_f32<<<grid, block, 0, stream>>>(img, ker, out);
}